// GCNModel_2_20504173871439
// MI455X (gfx1250) — compile-verified
//
#include <hip/hip_runtime.h>
#include <math.h>

#define HIDC 128

typedef __attribute__((ext_vector_type(2))) float v2f;
typedef __attribute__((ext_vector_type(8))) float v8f;

// ---------------------------------------------------------------------------
// Degree / normalization kernels
// ---------------------------------------------------------------------------
__global__ void init_deg_kernel(float* __restrict__ deg, int n) {
    int i = blockIdx.x * blockDim.x + threadIdx.x;
    if (i < n) deg[i] = 1.0f;  // self-loop contributes 1 to every node's degree
}

__global__ void count_deg_kernel(const int* __restrict__ col, float* deg, int e) {
    int i = blockIdx.x * blockDim.x + threadIdx.x;
    if (i < e) atomicAdd(&deg[col[i]], 1.0f);
}

__global__ void rsqrt_kernel(float* deg, int n) {
    int i = blockIdx.x * blockDim.x + threadIdx.x;
    if (i < n) deg[i] = rsqrtf(deg[i]);  // deg >= 1 always (self loops)
}

// ---------------------------------------------------------------------------
// Dense GEMM: C[n,128] = (optional relu)(A[n,K]) @ W[K,128] using
// V_WMMA_F32_16X16X4_F32. One wave computes a 16x128 strip. Weights in LDS.
// ---------------------------------------------------------------------------
template <int K, bool RELU_A>
__global__ __launch_bounds__(256) void gemm_wmma_kernel(
    const float* __restrict__ A, const float* __restrict__ W,
    float* __restrict__ C, int nrows) {
    __shared__ float sW[K * HIDC];
    for (int idx = threadIdx.x; idx < K * HIDC; idx += 256) sW[idx] = W[idx];
    __syncthreads();

    int wave = threadIdx.x >> 5;
    int rowblock = blockIdx.x * 8 + wave;
    int row0 = rowblock * 16;
    if (row0 >= nrows) return;

    int lane = threadIdx.x & 31;
    int half = lane >> 4;   // 0: K pair {0,1} of the 4-slice, 1: pair {2,3}
    int lm   = lane & 15;   // row (A) / column (B,D) within the 16-wide tile

    v8f acc[8];
#pragma unroll
    for (int t = 0; t < 8; ++t) acc[t] = (v8f)(0.0f);

    int arow_idx = row0 + lm;
    if (arow_idx >= nrows) arow_idx = nrows - 1;  // clamp (partial tail tile)
    const float* arow = A + (size_t)arow_idx * K;

#pragma unroll 4
    for (int ks = 0; ks < K / 4; ++ks) {
        int k0 = ks * 4 + half * 2;
        float a0 = arow[k0];
        float a1 = arow[k0 + 1];
        if (RELU_A) { a0 = fmaxf(a0, 0.0f); a1 = fmaxf(a1, 0.0f); }
        v2f av; av.x = a0; av.y = a1;
#pragma unroll
        for (int t = 0; t < 8; ++t) {
            v2f bv;
            bv.x = sW[(k0    ) * HIDC + t * 16 + lm];
            bv.y = sW[(k0 + 1) * HIDC + t * 16 + lm];
            acc[t] = __builtin_amdgcn_wmma_f32_16x16x4_f32(
                false, av, false, bv, (short)0, acc[t], false, false);
        }
    }

#pragma unroll
    for (int t = 0; t < 8; ++t) {
#pragma unroll
        for (int r = 0; r < 8; ++r) {
            int row = row0 + half * 8 + r;
            if (row < nrows)
                C[(size_t)row * HIDC + t * 16 + lm] = acc[t][r];
        }
    }
}

// ---------------------------------------------------------------------------
// agg[i,c] = bias[c] + h[i,c] * dinv[i]^2   (self-loop term + bias, fused)
// ---------------------------------------------------------------------------
__global__ void init_agg_kernel(const float* __restrict__ h,
                                const float* __restrict__ dinv,
                                const float* __restrict__ bias,
                                float* __restrict__ agg, int n) {
    int idx = blockIdx.x * blockDim.x + threadIdx.x;
    if (idx < n * HIDC) {
        int i = idx >> 7;
        int c = idx & (HIDC - 1);
        float d = dinv[i];
        agg[idx] = bias[c] + h[idx] * d * d;
    }
}

// ---------------------------------------------------------------------------
// Edge scatter: agg[col] += h[row] * dinv[row]*dinv[col]
// 32 threads per edge, 4 channels each -> coalesced 512B gather per edge.
// ---------------------------------------------------------------------------
__global__ __launch_bounds__(256) void scatter_kernel(
    const float* __restrict__ h, const float* __restrict__ dinv,
    const int* __restrict__ row, const int* __restrict__ col,
    float* __restrict__ agg, int e) {
    long long idx = (long long)blockIdx.x * blockDim.x + threadIdx.x;
    if (idx >= (long long)e * 32) return;
    int eidx = (int)(idx >> 5);
    int c4 = ((int)idx & 31) * 4;
    int r = row[eidx];
    int c = col[eidx];
    float w = dinv[r] * dinv[c];
    const float4 v = *(const float4*)(h + (size_t)r * HIDC + c4);
    float* ap = agg + (size_t)c * HIDC + c4;
    atomicAdd(ap + 0, v.x * w);
    atomicAdd(ap + 1, v.y * w);
    atomicAdd(ap + 2, v.z * w);
    atomicAdd(ap + 3, v.w * w);
}

// ---------------------------------------------------------------------------
// Fused per-node head: gg / a1 / a2 branches + final MLP + sigmoid.
// All weights (~44 KB) staged in LDS, one thread per node.
// ---------------------------------------------------------------------------
__global__ __launch_bounds__(256) void final_fused_kernel(
    const float* __restrict__ x, const float* __restrict__ agg1,
    const float* __restrict__ agg2,
    const float* ln1_w, const float* ln1_b, const float* ln2_w, const float* ln2_b,
    const float* na1_w, const float* na1_b, const float* na2_w, const float* na2_b,
    const float* na3_w, const float* na3_b, const float* na4_w, const float* na4_b,
    const float* f1_w, const float* f1_b, const float* f2_w, const float* f2_b,
    const float* f3_w, const float* f3_b, float* __restrict__ out, int n) {
    __shared__ float s[10993];
    float* s_ln1w = s;          // 16*32 = 512
    float* s_ln1b = s + 512;    // 32
    float* s_ln2w = s + 544;    // 32*16 = 512
    float* s_ln2b = s + 1056;   // 16
    float* s_na1w = s + 1072;   // 128*16 = 2048
    float* s_na1b = s + 3120;   // 16
    float* s_na2w = s + 3136;   // 16*16 = 256
    float* s_na2b = s + 3392;   // 16
    float* s_na3w = s + 3408;   // 2048
    float* s_na3b = s + 5456;   // 16
    float* s_na4w = s + 5472;   // 256
    float* s_na4b = s + 5728;   // 16
    float* s_f1w  = s + 5744;   // 48*64 = 3072
    float* s_f1b  = s + 8816;   // 64
    float* s_f2w  = s + 8880;   // 64*32 = 2048
    float* s_f2b  = s + 10928;  // 32
    float* s_f3w  = s + 10960;  // 32
    float* s_f3b  = s + 10992;  // 1

    int t = threadIdx.x;
    auto cp = [&](float* dst, const float* src, int cnt) {
        for (int k = t; k < cnt; k += 256) dst[k] = src[k];
    };
    cp(s_ln1w, ln1_w, 512);  cp(s_ln1b, ln1_b, 32);
    cp(s_ln2w, ln2_w, 512);  cp(s_ln2b, ln2_b, 16);
    cp(s_na1w, na1_w, 2048); cp(s_na1b, na1_b, 16);
    cp(s_na2w, na2_w, 256);  cp(s_na2b, na2_b, 16);
    cp(s_na3w, na3_w, 2048); cp(s_na3b, na3_b, 16);
    cp(s_na4w, na4_w, 256);  cp(s_na4b, na4_b, 16);
    cp(s_f1w, f1_w, 3072);   cp(s_f1b, f1_b, 64);
    cp(s_f2w, f2_w, 2048);   cp(s_f2b, f2_b, 32);
    cp(s_f3w, f3_w, 32);     cp(s_f3b, f3_b, 1);
    __syncthreads();

    int i = blockIdx.x * blockDim.x + threadIdx.x;
    if (i >= n) return;

    float xf[48];  // [gg(16) | a1(16) | a2(16)]

    // --- gg branch: relu(relu(x @ ln1 + b) @ ln2 + b) ---
    {
        float xin[16];
#pragma unroll
        for (int c = 0; c < 16; ++c) xin[c] = x[(size_t)i * 16 + c];
        float t32[32];
#pragma unroll
        for (int o = 0; o < 32; ++o) {
            float sacc = s_ln1b[o];
#pragma unroll
            for (int c = 0; c < 16; ++c) sacc = fmaf(xin[c], s_ln1w[c * 32 + o], sacc);
            t32[o] = fmaxf(sacc, 0.0f);
        }
#pragma unroll
        for (int o = 0; o < 16; ++o) {
            float sacc = s_ln2b[o];
#pragma unroll
            for (int c = 0; c < 32; ++c) sacc = fmaf(t32[c], s_ln2w[c * 16 + o], sacc);
            xf[o] = fmaxf(sacc, 0.0f);
        }
    }

    // --- a1 branch: relu(relu(x1 @ na1 + b) @ na2 + b), x1 = relu(agg1) ---
    {
        float t16[16];
#pragma unroll
        for (int o = 0; o < 16; ++o) t16[o] = s_na1b[o];
        const float* a1p = agg1 + (size_t)i * HIDC;
        for (int c = 0; c < HIDC; ++c) {
            float v = fmaxf(a1p[c], 0.0f);
#pragma unroll
            for (int o = 0; o < 16; ++o) t16[o] = fmaf(v, s_na1w[c * 16 + o], t16[o]);
        }
#pragma unroll
        for (int o = 0; o < 16; ++o) t16[o] = fmaxf(t16[o], 0.0f);
#pragma unroll
        for (int o = 0; o < 16; ++o) {
            float sacc = s_na2b[o];
#pragma unroll
            for (int c = 0; c < 16; ++c) sacc = fmaf(t16[c], s_na2w[c * 16 + o], sacc);
            xf[16 + o] = fmaxf(sacc, 0.0f);
        }
    }

    // --- a2 branch: relu(relu(x2 @ na3 + b) @ na4 + b), x2 = relu(agg2) ---
    {
        float t16[16];
#pragma unroll
        for (int o = 0; o < 16; ++o) t16[o] = s_na3b[o];
        const float* a2p = agg2 + (size_t)i * HIDC;
        for (int c = 0; c < HIDC; ++c) {
            float v = fmaxf(a2p[c], 0.0f);
#pragma unroll
            for (int o = 0; o < 16; ++o) t16[o] = fmaf(v, s_na3w[c * 16 + o], t16[o]);
        }
#pragma unroll
        for (int o = 0; o < 16; ++o) t16[o] = fmaxf(t16[o], 0.0f);
#pragma unroll
        for (int o = 0; o < 16; ++o) {
            float sacc = s_na4b[o];
#pragma unroll
            for (int c = 0; c < 16; ++c) sacc = fmaf(t16[c], s_na4w[c * 16 + o], sacc);
            xf[32 + o] = fmaxf(sacc, 0.0f);
        }
    }

    // --- final MLP: 48 -> 64 -> 32 -> 1, sigmoid ---
    float h64[64];
#pragma unroll
    for (int o = 0; o < 64; ++o) {
        float sacc = s_f1b[o];
#pragma unroll
        for (int c = 0; c < 48; ++c) sacc = fmaf(xf[c], s_f1w[c * 64 + o], sacc);
        h64[o] = fmaxf(sacc, 0.0f);
    }
    float h32[32];
#pragma unroll
    for (int o = 0; o < 32; ++o) {
        float sacc = s_f2b[o];
#pragma unroll
        for (int c = 0; c < 64; ++c) sacc = fmaf(h64[c], s_f2w[c * 32 + o], sacc);
        h32[o] = fmaxf(sacc, 0.0f);
    }
    float z = s_f3b[0];
#pragma unroll
    for (int c = 0; c < 32; ++c) z = fmaf(h32[c], s_f3w[c], z);
    out[i] = 1.0f / (1.0f + expf(-z));
}

// ---------------------------------------------------------------------------
// Launch
// ---------------------------------------------------------------------------
extern "C" void kernel_launch(void* const* d_in, const int* in_sizes, int n_in,
                              void* d_out, int out_size, void* d_ws, size_t ws_size,
                              hipStream_t stream) {
    const float* x       = (const float*)d_in[0];
    const int*   edge    = (const int*)d_in[1];
    const float* conv1_w = (const float*)d_in[2];
    const float* conv1_b = (const float*)d_in[3];
    const float* conv2_w = (const float*)d_in[4];
    const float* conv2_b = (const float*)d_in[5];
    const float* ln1_w = (const float*)d_in[6];
    const float* ln1_b = (const float*)d_in[7];
    const float* ln2_w = (const float*)d_in[8];
    const float* ln2_b = (const float*)d_in[9];
    const float* na1_w = (const float*)d_in[10];
    const float* na1_b = (const float*)d_in[11];
    const float* na2_w = (const float*)d_in[12];
    const float* na2_b = (const float*)d_in[13];
    const float* na3_w = (const float*)d_in[14];
    const float* na3_b = (const float*)d_in[15];
    const float* na4_w = (const float*)d_in[16];
    const float* na4_b = (const float*)d_in[17];
    const float* f1_w = (const float*)d_in[18];
    const float* f1_b = (const float*)d_in[19];
    const float* f2_w = (const float*)d_in[20];
    const float* f2_b = (const float*)d_in[21];
    const float* f3_w = (const float*)d_in[22];
    const float* f3_b = (const float*)d_in[23];
    float* out = (float*)d_out;

    int n = in_sizes[0] / 16;  // N nodes
    int e = in_sizes[1] / 2;   // E edges
    const int* erow = edge;      // edge_index[0] = source
    const int* ecol = edge + e;  // edge_index[1] = target

    float* ws = (float*)d_ws;
    size_t off = 0;
    float* dinv = ws + off; off += ((size_t)n + 255) & ~(size_t)255;
    float* h1   = ws + off; off += (size_t)n * HIDC;
    float* agg1 = ws + off; off += (size_t)n * HIDC;
    float* h2   = ws + off; off += (size_t)n * HIDC;
    float* agg2 = ws + off; off += (size_t)n * HIDC;

    const int B = 256;
    int gN  = (n + B - 1) / B;
    int gE  = (e + B - 1) / B;
    int gNC = (int)(((size_t)n * HIDC + B - 1) / B);
    int gE32 = (int)(((long long)e * 32 + B - 1) / B);
    int rowblocks = (n + 15) / 16;
    int gG = (rowblocks + 7) / 8;  // 8 waves per block, one 16-row strip each

    // normalization: deg -> dinv
    init_deg_kernel<<<gN, B, 0, stream>>>(dinv, n);
    count_deg_kernel<<<gE, B, 0, stream>>>(ecol, dinv, e);
    rsqrt_kernel<<<gN, B, 0, stream>>>(dinv, n);

    // layer 1
    gemm_wmma_kernel<16, false><<<gG, B, 0, stream>>>(x, conv1_w, h1, n);
    init_agg_kernel<<<gNC, B, 0, stream>>>(h1, dinv, conv1_b, agg1, n);
    scatter_kernel<<<gE32, B, 0, stream>>>(h1, dinv, erow, ecol, agg1, e);

    // layer 2 (ReLU of layer-1 output folded into the A-matrix load)
    gemm_wmma_kernel<128, true><<<gG, B, 0, stream>>>(agg1, conv2_w, h2, n);
    init_agg_kernel<<<gNC, B, 0, stream>>>(h2, dinv, conv2_b, agg2, n);
    scatter_kernel<<<gE32, B, 0, stream>>>(h2, dinv, erow, ecol, agg2, e);

    // fused per-node head
    final_fused_kernel<<<gN, B, 0, stream>>>(
        x, agg1, agg2,
        ln1_w, ln1_b, ln2_w, ln2_b,
        na1_w, na1_b, na2_w, na2_b,
        na3_w, na3_b, na4_w, na4_b,
        f1_w, f1_b, f2_w, f2_b, f3_w, f3_b, out, n);
}